// GraphBasedRegularization_81003083202619
// MI455X (gfx1250) — compile-verified
//
#include <hip/hip_runtime.h>
#include <hip/hip_bf16.h>

typedef float v2f __attribute__((ext_vector_type(2)));
typedef float v8f __attribute__((ext_vector_type(8)));

#define N_PTS   8192
#define DIM     128
#define NCLS    10
#define KNN     10
#define NSPLIT  8
#define BITW    (N_PTS / 32)      // 256 words per bit-row
#define LDS_STR 132               // padded stride: 132 % 64 = 4 -> conflict-free
#define ALPHA   0.99f

#if defined(__has_builtin)
#if __has_builtin(__builtin_amdgcn_global_load_async_to_lds_b128) && \
    __has_builtin(__builtin_amdgcn_s_wait_asynccnt)
#define HAVE_ASYNC_LDS 1
#endif
#endif

#if HAVE_ASYNC_LDS
// payload type per clang diagnostic: 'int __attribute__((__vector_size__(16)))'
typedef int b128_t __attribute__((vector_size(16)));
__device__ __forceinline__ void async_b128(const float* g, float* l) {
    __builtin_amdgcn_global_load_async_to_lds_b128(
        (__attribute__((address_space(1))) b128_t*)(uintptr_t)g,
        (__attribute__((address_space(3))) b128_t*)(uintptr_t)l,
        0, 0);
}
#endif

// ---------------- zero init (bits + flags) ----------------
__global__ void zero_kernel(unsigned* bits, int* flags) {
    int idx = blockIdx.x * blockDim.x + threadIdx.x;
    if (idx < N_PTS * BITW) bits[idx] = 0u;
    if (idx == 0) { flags[0] = 0; flags[1] = 0; }
}

// ---------------- row squared norms: one wave per row ----------------
__global__ void sqnorm_kernel(const float* __restrict__ X, float* __restrict__ sq) {
    int wave = (blockIdx.x * blockDim.x + threadIdx.x) >> 5;
    int lane = threadIdx.x & 31;
    if (wave >= N_PTS) return;
    const float4* row = (const float4*)(X + wave * DIM);
    float4 v = row[lane];                       // 32 lanes x 4 = 128
    float s = v.x * v.x + v.y * v.y + v.z * v.z + v.w * v.w;
    #pragma unroll
    for (int off = 16; off; off >>= 1) s += __shfl_down(s, off, 32);
    if (lane == 0) sq[wave] = s;
}

// lexicographic (d, idx) insert into ascending sorted top-10 (worst at [9])
__device__ __forceinline__ void topk_insert(float d2, int j, float* td, int* ti) {
    bool better = (d2 < td[9]) || (d2 == td[9] && j < ti[9]);
    if (better) {
        td[9] = d2; ti[9] = j;
        #pragma unroll
        for (int q = 9; q > 0; --q) {
            bool sw = (td[q] < td[q-1]) || (td[q] == td[q-1] && ti[q] < ti[q-1]);
            if (sw) {
                float t = td[q]; td[q] = td[q-1]; td[q-1] = t;
                int   u = ti[q]; ti[q] = ti[q-1]; ti[q-1] = u;
            }
        }
    }
}

// stage a 16x128 f32 tile (rows [base, base+16) of X) into padded LDS.
// Async path: GLOBAL_LOAD_ASYNC_TO_LDS_B128 (no VGPR staging, ASYNCcnt-tracked).
__device__ __forceinline__ void stage_tile(const float* __restrict__ X, int base,
                                           float* lds, int lane) {
    #pragma unroll
    for (int t = lane; t < 512; t += 32) {      // 512 float4 per tile
        int r = t >> 5, c4 = t & 31;
        const float* g = X + (base + r) * DIM + c4 * 4;
        float* l = lds + r * LDS_STR + c4 * 4;
#if HAVE_ASYNC_LDS
        async_b128(g, l);
#else
        *(float4*)l = *(const float4*)g;
#endif
    }
}

__device__ __forceinline__ void wait_stage() {
#if HAVE_ASYNC_LDS
    __builtin_amdgcn_s_wait_asynccnt(0);
#endif
    __syncthreads();
}

// ---------------- fused WMMA distance + per-row top-10 ----------------
// grid: (512 row-tiles, 8 column splits), block: 32 (one wave)
// B tiles double-buffered: async prefetch of tile ct+1 overlaps WMMA on tile ct.
__global__ __launch_bounds__(32)
void knn_dist_kernel(const float* __restrict__ X, const float* __restrict__ sq,
                     float* __restrict__ pd, int* __restrict__ pi) {
    __shared__ float ldsA[16 * LDS_STR];
    __shared__ float ldsB[2][16 * LDS_STR];
    __shared__ float ldsD[16 * 17];

    int lane = threadIdx.x;
    int rowBase  = blockIdx.x * 16;
    int split    = blockIdx.y;
    int colBase0 = split * (N_PTS / NSPLIT);
    const int NT = (N_PTS / NSPLIT) / 16;       // 64 column tiles per split

    // stage A tile + first B tile (both async), then wait once
    stage_tile(X, rowBase, ldsA, lane);
    stage_tile(X, colBase0, ldsB[0], lane);
    wait_stage();

    int l16 = lane & 15;
    int hi  = lane >> 4;   // 0: K=k,k+1 / M=0..7 half, 1: K=k+2,k+3 / M=8..15 half

    float td[KNN]; int ti[KNN];
    #pragma unroll
    for (int t = 0; t < KNN; ++t) { td[t] = 3.4e38f; ti[t] = 0x7fffffff; }

    float sqRow = sq[rowBase + l16];

    int buf = 0;
    for (int ct = 0; ct < NT; ++ct) {
        int colBase = colBase0 + ct * 16;
        // prefetch next B tile into the other buffer (overlaps WMMA below)
        if (ct + 1 < NT) stage_tile(X, colBase + 16, ldsB[buf ^ 1], lane);

        // 16x16 Gram tile via f32 WMMA, K = 128 in steps of 4
        const float* B = ldsB[buf];
        v8f cacc = {0.f, 0.f, 0.f, 0.f, 0.f, 0.f, 0.f, 0.f};
        #pragma unroll
        for (int kk = 0; kk < DIM; kk += 4) {
            v2f a, b;
            a.x = ldsA[l16 * LDS_STR + kk + 2 * hi];
            a.y = ldsA[l16 * LDS_STR + kk + 2 * hi + 1];
            b.x = B[l16 * LDS_STR + kk + 2 * hi];
            b.y = B[l16 * LDS_STR + kk + 2 * hi + 1];
            cacc = __builtin_amdgcn_wmma_f32_16x16x4_f32(
                false, a, false, b, (short)0, cacc, false, false);
        }
        // spill D: lane half hi owns rows M = hi*8 + r at N = l16
        #pragma unroll
        for (int r = 0; r < 8; ++r)
            ldsD[(hi * 8 + r) * 17 + l16] = cacc[r];
        __syncthreads();

        // scan: lane pair (l16, l16+16) covers row l16; halves of 16 columns
        #pragma unroll
        for (int t = 0; t < 8; ++t) {
            int n = hi * 8 + t;
            float g  = ldsD[l16 * 17 + n];
            int   j  = colBase + n;
            float d2 = sqRow - 2.0f * g + sq[j];
            topk_insert(d2, j, td, ti);
        }
        // ensure prefetch landed + ldsD reads done before next tile reuses them
        wait_stage();
        buf ^= 1;
    }

    // write partial top-10 lists: layout [row][split][half][10]
    int grow = rowBase + l16;
    long base = (long)grow * (NSPLIT * 2 * KNN) + (split * 2 + hi) * KNN;
    #pragma unroll
    for (int t = 0; t < KNN; ++t) { pd[base + t] = td[t]; pi[base + t] = ti[t]; }
}

// ---------------- merge 160 partial candidates/row -> symmetric bitmask ----------------
__global__ void knn_merge_kernel(const float* __restrict__ pd, const int* __restrict__ pi,
                                 unsigned* __restrict__ bits) {
    int row = blockIdx.x * blockDim.x + threadIdx.x;
    if (row >= N_PTS) return;
    float td[KNN]; int ti[KNN];
    #pragma unroll
    for (int t = 0; t < KNN; ++t) { td[t] = 3.4e38f; ti[t] = 0x7fffffff; }
    long base = (long)row * (NSPLIT * 2 * KNN);
    for (int s = 0; s < NSPLIT * 2 * KNN; ++s)
        topk_insert(pd[base + s], pi[base + s], td, ti);
    #pragma unroll
    for (int t = 0; t < KNN; ++t) {
        int j = ti[t];
        atomicOr(&bits[row * BITW + (j >> 5)], 1u << (j & 31));   // W[row, j]
        atomicOr(&bits[j * BITW + (row >> 5)], 1u << (row & 31)); // W[j, row]
    }
}

// ---------------- degrees ----------------
__global__ void degree_kernel(const unsigned* __restrict__ bits,
                              float* __restrict__ deg, float* __restrict__ invdeg) {
    int row = blockIdx.x * blockDim.x + threadIdx.x;
    if (row >= N_PTS) return;
    int cnt = 0;
    for (int w = 0; w < BITW; ++w) cnt += __popc(bits[row * BITW + w]);
    float d = (float)cnt;
    deg[row] = d;
    invdeg[row] = cnt > 0 ? 1.0f / d : 0.0f;
}

// ---------------- one-hot F0 ----------------
__global__ void f0_kernel(const int* __restrict__ labels, float* __restrict__ F) {
    int idx = blockIdx.x * blockDim.x + threadIdx.x;
    if (idx >= N_PTS * NCLS) return;
    int i = idx / NCLS, c = idx - i * NCLS;
    F[idx] = (labels[i] == c) ? 1.0f : 0.0f;
}

// ---------------- SpMM over bitmask: dst = alpha*deg_i*sum_j W_ij*invdeg_j*src_j + beta*src_i
// one wave per row; lane scans 8 bit-words
__global__ void spmm_kernel(const unsigned* __restrict__ bits,
                            const float* __restrict__ deg, const float* __restrict__ invdeg,
                            const float* __restrict__ src, float* __restrict__ dst,
                            float beta, int check, const int* __restrict__ done,
                            int* __restrict__ notclose) {
    if (check && *done) return;
    int wave = (blockIdx.x * blockDim.x + threadIdx.x) >> 5;
    int lane = threadIdx.x & 31;
    if (wave >= N_PTS) return;
    int row = wave;
    float acc[NCLS];
    #pragma unroll
    for (int c = 0; c < NCLS; ++c) acc[c] = 0.0f;
    const unsigned* wrow = bits + row * BITW;
    #pragma unroll
    for (int w = 0; w < BITW / 32; ++w) {
        int wi = lane * (BITW / 32) + w;
        unsigned word = wrow[wi];
        while (word) {
            int b = __ffs(word) - 1;
            word &= word - 1;
            int j = wi * 32 + b;
            float s = invdeg[j];
            const float* fj = src + j * NCLS;
            #pragma unroll
            for (int c = 0; c < NCLS; ++c) acc[c] += s * fj[c];
        }
    }
    #pragma unroll
    for (int off = 16; off; off >>= 1) {
        #pragma unroll
        for (int c = 0; c < NCLS; ++c) acc[c] += __shfl_down(acc[c], off, 32);
    }
    if (lane == 0) {
        float dg = deg[row];
        bool bad = false;
        #pragma unroll
        for (int c = 0; c < NCLS; ++c) {
            float fold = src[row * NCLS + c];
            float v = ALPHA * dg * acc[c] + beta * fold;
            dst[row * NCLS + c] = v;
            bad |= (fabsf(v - fold) > 1e-6f + 1e-5f * fabsf(fold));
        }
        if (check && bad) atomicOr(notclose, 1);
    }
}

// commit F_new only if !done && !close  (torch breaks BEFORE F = F_new)
__global__ void commit_kernel(const int* __restrict__ done, const int* __restrict__ notclose,
                              const float* __restrict__ Ftmp, float* __restrict__ F) {
    if (*done || (*notclose == 0)) return;
    int idx = blockIdx.x * blockDim.x + threadIdx.x;
    if (idx < N_PTS * NCLS) F[idx] = Ftmp[idx];
}

__global__ void control_kernel(int* done, int* notclose) {
    if (!*done && *notclose == 0) *done = 1;
    *notclose = 0;
}

extern "C" void kernel_launch(void* const* d_in, const int* in_sizes, int n_in,
                              void* d_out, int out_size, void* d_ws, size_t ws_size,
                              hipStream_t stream) {
    const float* X      = (const float*)d_in[0];
    const int*   labels = (const int*)d_in[1];

    char* ws = (char*)d_ws;
    size_t off = 0;
    float*    sq     = (float*)(ws + off);    off += (size_t)N_PTS * 4;
    float*    deg    = (float*)(ws + off);    off += (size_t)N_PTS * 4;
    float*    invdeg = (float*)(ws + off);    off += (size_t)N_PTS * 4;
    unsigned* bits   = (unsigned*)(ws + off); off += (size_t)N_PTS * BITW * 4;
    float*    pd     = (float*)(ws + off);    off += (size_t)N_PTS * NSPLIT * 2 * KNN * 4;
    int*      pi     = (int*)(ws + off);      off += (size_t)N_PTS * NSPLIT * 2 * KNN * 4;
    float*    bufA   = (float*)(ws + off);    off += (size_t)N_PTS * NCLS * 4;   // F0 / F_tmp
    float*    bufB   = (float*)(ws + off);    off += (size_t)N_PTS * NCLS * 4;   // F (current)
    int*      flags  = (int*)(ws + off);      off += 256;                        // [done, notclose]
    if (off > ws_size) return;

    int* done     = flags + 0;
    int* notclose = flags + 1;

    // 1) init bits + flags (ws not re-poisoned between replays)
    zero_kernel<<<(N_PTS * BITW + 255) / 256, 256, 0, stream>>>(bits, flags);
    // 2) squared norms
    sqnorm_kernel<<<(N_PTS * 32 + 255) / 256, 256, 0, stream>>>(X, sq);
    // 3) WMMA distances + per-row top-10 partials (async double-buffered B)
    knn_dist_kernel<<<dim3(N_PTS / 16, NSPLIT), 32, 0, stream>>>(X, sq, pd, pi);
    // 4) merge partials -> symmetric adjacency bitmask
    knn_merge_kernel<<<(N_PTS + 255) / 256, 256, 0, stream>>>(pd, pi, bits);
    // 5) degrees
    degree_kernel<<<(N_PTS + 255) / 256, 256, 0, stream>>>(bits, deg, invdeg);
    // 6) F0 one-hot
    f0_kernel<<<(N_PTS * NCLS + 255) / 256, 256, 0, stream>>>(labels, bufA);
    // 7) F = F0 + alpha * S @ F0   (beta = 1, no close-check)
    spmm_kernel<<<(N_PTS * 32) / 256, 256, 0, stream>>>(bits, deg, invdeg, bufA, bufB,
                                                        1.0f, 0, done, notclose);
    // 8) 100 diffusion steps with early-freeze semantics
    for (int it = 0; it < 100; ++it) {
        spmm_kernel<<<(N_PTS * 32) / 256, 256, 0, stream>>>(bits, deg, invdeg, bufB, bufA,
                                                            1.0f - ALPHA, 1, done, notclose);
        commit_kernel<<<(N_PTS * NCLS + 255) / 256, 256, 0, stream>>>(done, notclose, bufA, bufB);
        control_kernel<<<1, 1, 0, stream>>>(done, notclose);
    }
    // 9) output
    (void)hipMemcpyAsync(d_out, bufB, (size_t)N_PTS * NCLS * 4,
                         hipMemcpyDeviceToDevice, stream);
}